// PolynormerNet_86930138071450
// MI455X (gfx1250) — compile-verified
//
#include <hip/hip_runtime.h>
#include <cstdint>
#include <cstddef>

typedef float v2f __attribute__((ext_vector_type(2)));
typedef float v8f __attribute__((ext_vector_type(8)));

#define NNODES 100000
#define CH 128
#define OUTC 40
#define NEDGE 3200000

// XOR-swizzled LDS index for transposed weight tile: logical [c][k], stride 128.
// Swizzle keeps bits[1:0] of k, so (k even, k+1) stay contiguous for float2 loads,
// and the two half-waves (k vs k+2) hit disjoint bank groups.
__device__ __forceinline__ int swz(int c, int k) {
  return c * 128 + (k ^ ((c & 15) << 2));
}

// ---------------- degree / norm ----------------
__global__ void deg_init_kernel(float* __restrict__ deg, int n) {
  int i = blockIdx.x * blockDim.x + threadIdx.x;
  if (i < n) deg[i] = 1.0f;  // self-loop contribution
}

__global__ void deg_edges_kernel(const long long* __restrict__ dst,
                                 float* __restrict__ deg, int ne) {
  int e = blockIdx.x * blockDim.x + threadIdx.x;
  if (e < ne) atomicAdd(&deg[dst[e]], 1.0f);
}

__global__ void dinv_kernel(float* __restrict__ deg, int n) {
  int i = blockIdx.x * blockDim.x + threadIdx.x;
  if (i < n) {
    float d = deg[i];
    deg[i] = (d > 0.0f) ? rsqrtf(d) : 0.0f;  // in place -> dinv
  }
}

// ---------------- dense 128x128 GEMM via WMMA f32 16x16x4 ----------------
// Y[r,c] = act( sum_k X[r,k]*W[k,c] + bias[c] )   (bias/relu only if fuse!=0)
__global__ __launch_bounds__(256) void gemm128_kernel(
    const float* __restrict__ X, const float* __restrict__ W,
    const float* __restrict__ bias, float* __restrict__ Y,
    int nrows, int fuse_bias_relu) {
  __shared__ float sW[CH * 128];  // transposed+swizzled: sW[swz(c,k)] = W[k*CH+c]

  const int tid = threadIdx.x;
  for (int idx = tid; idx < CH * CH; idx += 256) {
    int k = idx >> 7, c = idx & 127;
    sW[swz(c, k)] = W[idx];
  }
  __syncthreads();

  const int wave = tid >> 5;
  const int lane = tid & 31;
  const int m16 = lane & 15;    // M for A-frag, N for B/C-frags
  const int khalf = lane >> 4;  // 0/1 -> K offset 0/2
  const int rowBase = blockIdx.x * 128 + wave * 16;

  v8f acc[8] = {};

  int arow = rowBase + m16;
  if (arow >= nrows) arow = nrows - 1;  // clamp; OOB rows not stored
  const float* aptr = X + (size_t)arow * CH;

  for (int kk = 0; kk < CH; kk += 4) {
    const int kb = kk + 2 * khalf;
    v2f a = *(const v2f*)(aptr + kb);
#pragma unroll
    for (int n = 0; n < 8; ++n) {
      const int col = n * 16 + m16;
      v2f b = *(const v2f*)&sW[swz(col, kb)];
      acc[n] = __builtin_amdgcn_wmma_f32_16x16x4_f32(
          false, a, false, b, (short)0, acc[n], false, false);
    }
  }

#pragma unroll
  for (int n = 0; n < 8; ++n) {
    const int col = n * 16 + m16;
    const float bc = fuse_bias_relu ? bias[col] : 0.0f;
#pragma unroll
    for (int j = 0; j < 8; ++j) {
      const int row = rowBase + j + 8 * khalf;
      if (row < nrows) {
        float v = acc[n][j];
        if (fuse_bias_relu) v = fmaxf(v + bc, 0.0f);
        Y[(size_t)row * CH + col] = v;
      }
    }
  }
}

// ---------------- 128 -> 40 GEMM (padded to 48 cols), bias, no act ----------
__global__ __launch_bounds__(256) void gemm_out_kernel(
    const float* __restrict__ X, const float* __restrict__ W,  // [CH, OUTC]
    const float* __restrict__ bias, float* __restrict__ Y,     // [nrows, OUTC]
    int nrows) {
  __shared__ float sW[48 * 128];

  const int tid = threadIdx.x;
  for (int idx = tid; idx < CH * 48; idx += 256) {
    int k = idx / 48, c = idx % 48;
    sW[swz(c, k)] = (c < OUTC) ? W[k * OUTC + c] : 0.0f;
  }
  __syncthreads();

  const int wave = tid >> 5;
  const int lane = tid & 31;
  const int m16 = lane & 15;
  const int khalf = lane >> 4;
  const int rowBase = blockIdx.x * 128 + wave * 16;

  v8f acc[3] = {};

  int arow = rowBase + m16;
  if (arow >= nrows) arow = nrows - 1;
  const float* aptr = X + (size_t)arow * CH;

  for (int kk = 0; kk < CH; kk += 4) {
    const int kb = kk + 2 * khalf;
    v2f a = *(const v2f*)(aptr + kb);
#pragma unroll
    for (int n = 0; n < 3; ++n) {
      const int col = n * 16 + m16;
      v2f b = *(const v2f*)&sW[swz(col, kb)];
      acc[n] = __builtin_amdgcn_wmma_f32_16x16x4_f32(
          false, a, false, b, (short)0, acc[n], false, false);
    }
  }

#pragma unroll
  for (int n = 0; n < 3; ++n) {
    const int col = n * 16 + m16;
    if (col >= OUTC) continue;
    const float bc = bias[col];
#pragma unroll
    for (int j = 0; j < 8; ++j) {
      const int row = rowBase + j + 8 * khalf;
      if (row < nrows) Y[(size_t)row * OUTC + col] = acc[n][j] + bc;
    }
  }
}

// ---------------- edge scatter: agg[dst] += t[src] * dinv[s]*dinv[d] --------
__global__ __launch_bounds__(256) void scatter_kernel(
    const float* __restrict__ t, const long long* __restrict__ src,
    const long long* __restrict__ dst, const float* __restrict__ dinv,
    float* __restrict__ agg, int nedges) {
  const long long gid = (long long)blockIdx.x * blockDim.x + threadIdx.x;
  const int e = (int)(gid >> 5);
  const int lane = threadIdx.x & 31;
  if (e >= nedges) return;
  const long long s = src[e];
  const long long d = dst[e];
  const float w = dinv[s] * dinv[d];
  const float4 v = ((const float4*)(t + s * CH))[lane];
  float* o = agg + d * CH + (size_t)lane * 4;
  atomicAdd(o + 0, v.x * w);
  atomicAdd(o + 1, v.y * w);
  atomicAdd(o + 2, v.z * w);
  atomicAdd(o + 3, v.w * w);
}

// ---------------- zero buffer ----------------
__global__ void zero4_kernel(float4* __restrict__ p, long long n4) {
  long long i = (long long)blockIdx.x * blockDim.x + threadIdx.x;
  if (i < n4) p[i] = make_float4(0.f, 0.f, 0.f, 0.f);
}

// -------- epilogue: h = relu(agg + t*dinv^2 (self loop) + bias) -------------
__global__ void finish_conv_kernel(const float* __restrict__ agg,
                                   const float* __restrict__ t,
                                   const float* __restrict__ dinv,
                                   const float* __restrict__ bias,
                                   float* __restrict__ h, int nrows) {
  long long i = (long long)blockIdx.x * blockDim.x + threadIdx.x;  // float4 idx
  long long n4 = (long long)nrows * (CH / 4);
  if (i >= n4) return;
  const int node = (int)(i >> 5);
  const int q = (int)(i & 31);
  const float di = dinv[node];
  const float w = di * di;
  const float4 a = ((const float4*)agg)[i];
  const float4 tv = ((const float4*)t)[i];
  const int c0 = q * 4;
  float4 r;
  r.x = fmaxf(a.x + tv.x * w + bias[c0 + 0], 0.0f);
  r.y = fmaxf(a.y + tv.y * w + bias[c0 + 1], 0.0f);
  r.z = fmaxf(a.z + tv.z * w + bias[c0 + 2], 0.0f);
  r.w = fmaxf(a.w + tv.w * w + bias[c0 + 3], 0.0f);
  ((float4*)h)[i] = r;
}

// ---------------- in-place log-softmax over 40 cols ----------------
__global__ void logsoftmax_kernel(float* __restrict__ Y, int nrows) {
  int r = blockIdx.x * blockDim.x + threadIdx.x;
  if (r >= nrows) return;
  float* p = Y + (size_t)r * OUTC;
  float v[OUTC];
  float m = -INFINITY;
#pragma unroll
  for (int c = 0; c < OUTC; ++c) { v[c] = p[c]; m = fmaxf(m, v[c]); }
  float s = 0.0f;
#pragma unroll
  for (int c = 0; c < OUTC; ++c) s += expf(v[c] - m);
  const float ls = logf(s) + m;
#pragma unroll
  for (int c = 0; c < OUTC; ++c) p[c] = v[c] - m - ls + 0.0f * 0.0f, p[c] = v[c] - (m + ls - m);  // p[c] = v[c] - (log(sum)+m)
}

extern "C" void kernel_launch(void* const* d_in, const int* in_sizes, int n_in,
                              void* d_out, int out_size, void* d_ws, size_t ws_size,
                              hipStream_t stream) {
  const float* x = (const float*)d_in[0];
  const long long* ei = (const long long*)d_in[1];
  const float* lin1_w = (const float*)d_in[2];
  const float* lin1_b = (const float*)d_in[3];
  const float* gcn_w = (const float*)d_in[4];   // [3,128,128]
  const float* gcn_b = (const float*)d_in[5];   // [3,128]
  const float* lin2_w = (const float*)d_in[6];  // [128,40]
  const float* lin2_b = (const float*)d_in[7];  // [40]
  float* out = (float*)d_out;

  const long long* src = ei;
  const long long* dst = ei + NEDGE;

  char* ws = (char*)d_ws;
  const size_t featBytes = (size_t)NNODES * CH * sizeof(float);  // 51.2 MB
  float* h = (float*)(ws);
  float* t = (float*)(ws + featBytes);
  float* agg = (float*)(ws + 2 * featBytes);
  float* dinv = (float*)(ws + 3 * featBytes);  // N floats (deg -> dinv in place)

  const int threads = 256;
  const int nodeBlocks = (NNODES + threads - 1) / threads;
  const int edgeBlocks = (NEDGE + threads - 1) / threads;
  const int gemmBlocks = (NNODES + 127) / 128;
  const long long n4 = (long long)NNODES * (CH / 4);
  const int n4Blocks = (int)((n4 + threads - 1) / threads);
  const int scatBlocks = (int)(((long long)NEDGE * 32 + threads - 1) / threads);

  // degree -> dinv
  deg_init_kernel<<<nodeBlocks, threads, 0, stream>>>(dinv, NNODES);
  deg_edges_kernel<<<edgeBlocks, threads, 0, stream>>>(dst, dinv, NEDGE);
  dinv_kernel<<<nodeBlocks, threads, 0, stream>>>(dinv, NNODES);

  // h = relu(x @ W1 + b1)
  gemm128_kernel<<<gemmBlocks, threads, 0, stream>>>(x, lin1_w, lin1_b, h,
                                                     NNODES, 1);

  for (int k = 0; k < 3; ++k) {
    const float* Wk = gcn_w + (size_t)k * CH * CH;
    const float* bk = gcn_b + (size_t)k * CH;
    // t = h @ Wk
    gemm128_kernel<<<gemmBlocks, threads, 0, stream>>>(h, Wk, bk, t, NNODES, 0);
    // agg = 0; agg[dst] += t[src]*norm  (self loops folded into epilogue)
    zero4_kernel<<<n4Blocks, threads, 0, stream>>>((float4*)agg, n4);
    scatter_kernel<<<scatBlocks, threads, 0, stream>>>(t, src, dst, dinv, agg,
                                                       NEDGE);
    // h = relu(agg + t*dinv^2 + bk)
    finish_conv_kernel<<<n4Blocks, threads, 0, stream>>>(agg, t, dinv, bk, h,
                                                         NNODES);
  }

  // logits -> d_out, then in-place log_softmax
  gemm_out_kernel<<<gemmBlocks, threads, 0, stream>>>(h, lin2_w, lin2_b, out,
                                                      NNODES);
  logsoftmax_kernel<<<nodeBlocks, threads, 0, stream>>>(out, NNODES);
}